// _BaseMutualEmbeddingModel_64854006170403
// MI455X (gfx1250) — compile-verified
//
#include <hip/hip_runtime.h>
#include <hip/hip_bf16.h>

// t-SNE KL divergence on MI455X (gfx1250).
// Pass 1: S = sum_{i!=j} temp(i,j)                      (pure compute, WMMA Gram tiles)
// Pass 2: KL = sum_{P>0} P*log(P / max(temp/S, eps))    (streams P: 256MB @ HBM)
// Distances via V_WMMA_F32_16X16X4_F32 (the reference's mm-trick on the matrix pipe).

typedef __attribute__((ext_vector_type(2))) float v2f;
typedef __attribute__((ext_vector_type(8))) float v8f;

#define NN        8192
#define TILES     512                    // NN / 16
#define NBLK      512
#define THREADS   256
#define WAVES_PB  8                      // 256 / 32
#define TOT_WAVES (NBLK * WAVES_PB)      // 4096
#define TILES_PW  ((TILES * TILES) / TOT_WAVES)   // 64 tiles per wave; divides 512,
                                                  // so each wave stays on ONE tile-row
#define EPS_TSNE  1e-30f

template <bool WITH_P>
__global__ __launch_bounds__(THREADS)
void tsne_pass(const float* __restrict__ P,
               const float* __restrict__ Y,
               const float* __restrict__ Ssrc,
               float* __restrict__ partials)
{
    const int lane = threadIdx.x & 31;
    const int m    = lane & 15;          // A/B row id within tile; also column N in C
    const int half = lane >> 4;          // 0: K=0,1   1: K=2,3 (zero-padded, d=2)
    const int wave = blockIdx.x * WAVES_PB + (threadIdx.x >> 5);

    // Each wave owns one tile-row and 64 consecutive column-tiles within it.
    const int ti  = wave >> 3;           // 512 tile-rows, 8 waves per row
    const int tj0 = (wave & 7) << 6;     // starting column-tile (64 tiles each)
    const int i0  = ti << 4;

    const float2* __restrict__ Y2 = reinterpret_cast<const float2*>(Y);

    float invS = 0.0f;
    if (WITH_P) invS = __builtin_amdgcn_rcpf(*Ssrc);

    // ---- Loop-invariant A-side setup (hoisted: same ti for all 64 tiles) ----
    // A (16x4 f32) wave32 layout: lanes 0-15 hold row M=m, K=0,1 in vgpr0,1;
    // lanes 16-31 hold K=2,3 which are zero since d=2.
    const float2 yi  = Y2[i0 + m];       // 64KB Y stays L2/WGP$ resident
    const float  sqa = fmaf(yi.x, yi.x, yi.y * yi.y);   // valid in lanes 0-15
    v2f a;
    a.x = half ? 0.0f : yi.x;
    a.y = half ? 0.0f : yi.y;

    // sq_i for this lane's 8 output rows M = 8*half + k  (C-matrix layout)
    float sqi[8];
#pragma unroll
    for (int k = 0; k < 8; ++k)
        sqi[k] = __shfl(sqa, (half << 3) + k, 32);

    const int ibase = i0 + (half << 3);

    float acc = 0.0f;

    for (int tt = 0; tt < TILES_PW; ++tt) {
        const int tj = tj0 + tt;
        const int j0 = tj << 4;
        const int j  = j0 + m;

        // ---- B operand: (Y tile)^T has the same register image as A, built
        //      from rows j0+m ----
        const float2 yj  = Y2[j0 + m];
        const float  sqb = fmaf(yj.x, yj.x, yj.y * yj.y);
        v2f b;
        b.x = half ? 0.0f : yj.x;
        b.y = half ? 0.0f : yj.y;

        v8f c = {};
        // G[M][N] = Y_i . Y_j for the 16x16 tile, on the matrix pipe.
        v8f g = __builtin_amdgcn_wmma_f32_16x16x4_f32(
            /*neg_a=*/false, a, /*neg_b=*/false, b,
            /*c_mod=*/(short)0, c, /*reuse_a=*/false, /*reuse_b=*/false);

        const float sq_j = __shfl(sqb, m, 32);   // column N = m for this lane

        const float* __restrict__ Pbase = P + (size_t)ibase * NN + j;

        if (WITH_P) {
            // Keep all 16 P-row streams warm: lanes 0-15 prefetch +128 cols,
            // lanes 16-31 prefetch +256 cols (global_prefetch, speculative).
            int pcol = j0 + 128 + (half << 7);
            if (pcol > NN - 16) pcol = j0;
            __builtin_prefetch(P + (size_t)(i0 + m) * NN + pcol + m, 0, 1);
        }

        const bool diagTile = (!WITH_P) && (tj == ti);   // wave-uniform

#pragma unroll
        for (int k = 0; k < 8; ++k) {
            float d2 = fmaf(-2.0f, g[k], sqi[k] + sq_j);
            d2 = fmaxf(d2, 0.0f);
            const float pd   = (d2 > 0.0f) ? __builtin_amdgcn_sqrtf(d2) : 0.0f;
            const float t    = __builtin_amdgcn_rcpf(1.0f + pd);
            const float temp = t * (2.0f - t);

            if (WITH_P) {
                // 8 loads share one base VGPR pair; k*NN*4 <= 224KB fits IOFFSET.
                const float p = Pbase[(size_t)k * NN];
                if (p > 0.0f) {   // masked KL; diagonal has P==0 so it's excluded
                    const float q = fmaxf(temp * invS, EPS_TSNE);
                    acc += p * __logf(p * __builtin_amdgcn_rcpf(q));
                }
            } else if (diagTile) {
                if (ibase + k != j) acc += temp;   // fill_diagonal_(0)
            } else {
                acc += temp;
            }
        }
    }

    // ---- deterministic block reduction (no float atomics) ----
#pragma unroll
    for (int off = 16; off > 0; off >>= 1)
        acc += __shfl_down(acc, off, 32);

    __shared__ float sRed[WAVES_PB];
    const int wib = threadIdx.x >> 5;
    if (lane == 0) sRed[wib] = acc;
    __syncthreads();
    if (threadIdx.x == 0) {
        float s = 0.0f;
#pragma unroll
        for (int w = 0; w < WAVES_PB; ++w) s += sRed[w];
        partials[blockIdx.x] = s;
    }
}

// Reduce exactly 512 partials with one block of 256 threads (deterministic).
__global__ __launch_bounds__(256)
void reduce512(const float* __restrict__ in, float* __restrict__ out)
{
    __shared__ float sm[256];
    sm[threadIdx.x] = in[threadIdx.x] + in[threadIdx.x + 256];
    __syncthreads();
    for (int off = 128; off > 0; off >>= 1) {
        if (threadIdx.x < off) sm[threadIdx.x] += sm[threadIdx.x + off];
        __syncthreads();
    }
    if (threadIdx.x == 0) *out = sm[0];
}

extern "C" void kernel_launch(void* const* d_in, const int* in_sizes, int n_in,
                              void* d_out, int out_size, void* d_ws, size_t ws_size,
                              hipStream_t stream)
{
    const float* P = (const float*)d_in[0];   // [8192*8192] f32
    const float* Y = (const float*)d_in[1];   // [8192*2]    f32

    float* ws    = (float*)d_ws;
    float* part1 = ws;            // 512 partials for S
    float* Sdst  = ws + 512;      // reduced S
    float* part2 = ws + 1024;     // 512 partials for KL

    tsne_pass<false><<<NBLK, THREADS, 0, stream>>>(P, Y, Sdst, part1);
    reduce512<<<1, 256, 0, stream>>>(part1, Sdst);
    tsne_pass<true><<<NBLK, THREADS, 0, stream>>>(P, Y, Sdst, part2);
    reduce512<<<1, 256, 0, stream>>>(part2, (float*)d_out);
}